// MultiRelGCN_19413252178302
// MI455X (gfx1250) — compile-verified
//
#include <hip/hip_runtime.h>
#include <hip/hip_bf16.h>
#include <math.h>

typedef float v2f __attribute__((ext_vector_type(2)));
typedef float v8f __attribute__((ext_vector_type(8)));

#define NUM_USERS 100000
#define NUM_ITEMS 50000
#define EMB 64
#define NEDGE 1000000
#define BATCH 16384

// ---------------- elementwise helpers ----------------

__global__ void scale_f4(const float4* __restrict__ in, float4* __restrict__ out,
                         float s, int n4) {
    int i = blockIdx.x * blockDim.x + threadIdx.x;
    if (i < n4) {
        float4 v = in[i];
        out[i] = make_float4(v.x * s, v.y * s, v.z * s, v.w * s);
    }
}

__global__ void zero_f4(float4* __restrict__ p, int n4) {
    int i = blockIdx.x * blockDim.x + threadIdx.x;
    if (i < n4) p[i] = make_float4(0.f, 0.f, 0.f, 0.f);
}

// dst += s_t * (a + b), with s_t = softmax(tw)[t] / 3 computed in-kernel
__global__ void accum_layers(float4* __restrict__ dst,
                             const float4* __restrict__ a,
                             const float4* __restrict__ b,
                             const float* __restrict__ tw, int t, int n4) {
    int i = blockIdx.x * blockDim.x + threadIdx.x;
    if (i >= n4) return;
    float t0 = tw[0], t1 = tw[1];
    float m = fmaxf(t0, t1);
    float e0 = __expf(t0 - m), e1 = __expf(t1 - m);
    float s = ((t == 0) ? e0 : e1) / (e0 + e1) * (1.0f / 3.0f);
    float4 va = a[i], vb = b[i], vd = dst[i];
    vd.x += s * (va.x + vb.x);
    vd.y += s * (va.y + vb.y);
    vd.z += s * (va.z + vb.z);
    vd.w += s * (va.w + vb.w);
    dst[i] = vd;
}

// ---------------- edge scatter (segment_sum) ----------------
// One 32-lane wave per edge; each lane owns a float2 slice of the 64-dim row.
// i_out[dst] += w * u_in[src] ; u_out[src] += w * i_in[dst]
__global__ void scatter_layer(const float* __restrict__ u_in,
                              const float* __restrict__ i_in,
                              float* __restrict__ u_out,
                              float* __restrict__ i_out,
                              const int* __restrict__ edges,   // [2, NEDGE] flat
                              const float* __restrict__ w,
                              int num_edges) {
    int lane = threadIdx.x & 31;
    int grp  = blockIdx.x * (blockDim.x >> 5) + (threadIdx.x >> 5);
    int nGrp = gridDim.x * (blockDim.x >> 5);
    for (int e = grp; e < num_edges; e += nGrp) {
        int   s  = edges[e];
        int   d  = edges[num_edges + e];
        float wt = w[e];
        const float2* us = (const float2*)(u_in + (size_t)s * EMB);
        const float2* id = (const float2*)(i_in + (size_t)d * EMB);
        float2 uv = us[lane];
        float2 iv = id[lane];
        float* ip = i_out + (size_t)d * EMB + 2 * lane;
        float* up = u_out + (size_t)s * EMB + 2 * lane;
        unsafeAtomicAdd(ip + 0, wt * uv.x);   // -> global_atomic_add_f32 (no rtn)
        unsafeAtomicAdd(ip + 1, wt * uv.y);
        unsafeAtomicAdd(up + 0, wt * iv.x);
        unsafeAtomicAdd(up + 1, wt * iv.y);
    }
}

// ---------------- batched dot via fp32 WMMA ----------------
// One wave computes 16 dot products as diag(U(16x64) * I^T(64x16)) using
// sixteen V_WMMA_F32_16X16X4_F32 accumulation steps.
// A(16x4) lane layout: lanes 0-15 -> M=lane, K={0,1}; lanes 16-31 -> K={2,3}.
// B(4x16) lane layout mirrors it, so both operands are one float2 per lane.
__global__ void dot_wmma(const float* __restrict__ fU, const float* __restrict__ fI,
                         const int* __restrict__ uidx, const int* __restrict__ iidx,
                         float* __restrict__ out) {
    int lane = threadIdx.x & 31;
    int g    = blockIdx.x * (blockDim.x >> 5) + (threadIdx.x >> 5);
    int row  = lane & 15;
    int koff = (lane >> 4) << 1;     // 0 for lanes 0-15, 2 for lanes 16-31

    int ub = uidx[g * 16 + row];
    int ib = iidx[g * 16 + row];
    const float* up = fU + (size_t)ub * EMB + koff;
    const float* ip = fI + (size_t)ib * EMB + koff;

    v8f c = {};
#pragma unroll
    for (int k = 0; k < 16; ++k) {
        v2f a = *(const v2f*)(up + k * 4);
        v2f b = *(const v2f*)(ip + k * 4);
        c = __builtin_amdgcn_wmma_f32_16x16x4_f32(
                /*neg_a=*/false, a, /*neg_b=*/false, b,
                /*c_mod=*/(short)0, c, /*reuse_a=*/false, /*reuse_b=*/false);
    }

    // Diagonal extraction from the 16x16 f32 C layout:
    //   m<8  lives in lane m,     vgpr m
    //   m>=8 lives in lane m+16,  vgpr m-8
    int m = (lane < 8) ? lane : ((lane >= 24) ? (lane - 16) : -1);
    int r = (lane < 8) ? lane : (lane - 24);
    float dv = c[0];
#pragma unroll
    for (int j = 1; j < 8; ++j)
        if (r == j) dv = c[j];
    if (m >= 0) out[g * 16 + m] = dv;
}

// ---------------- launcher ----------------

extern "C" void kernel_launch(void* const* d_in, const int* in_sizes, int n_in,
                              void* d_out, int out_size, void* d_ws, size_t ws_size,
                              hipStream_t stream) {
    const int*   uidx = (const int*)  d_in[0];
    const int*   iidx = (const int*)  d_in[1];
    const int*   e0   = (const int*)  d_in[2];
    const float* w0   = (const float*)d_in[3];
    const int*   e1   = (const int*)  d_in[4];
    const float* w1   = (const float*)d_in[5];
    const float* uemb = (const float*)d_in[6];
    const float* iemb = (const float*)d_in[7];
    const float* tw   = (const float*)d_in[8];
    float*       out  = (float*)d_out;

    const size_t NU = (size_t)NUM_USERS * EMB;  // 6.4M floats
    const size_t NI = (size_t)NUM_ITEMS * EMB;  // 3.2M floats

    float* ws = (float*)d_ws;
    float* u1 = ws;            // layer-1 user output
    float* i1 = u1 + NU;       // layer-1 item output
    float* u2 = i1 + NI;       // layer-2 user output
    float* i2 = u2 + NU;       // layer-2 item output
    float* fU = i2 + NI;       // final user accumulator
    float* fI = fU + NU;       // final item accumulator

    const int tpb = 256;
    const int nu4 = (int)(NU / 4), ni4 = (int)(NI / 4);

    // finalU = user_emb/3 ; finalI = item_emb/3 (softmax weights sum to 1)
    scale_f4<<<(nu4 + tpb - 1) / tpb, tpb, 0, stream>>>(
        (const float4*)uemb, (float4*)fU, 1.0f / 3.0f, nu4);
    scale_f4<<<(ni4 + tpb - 1) / tpb, tpb, 0, stream>>>(
        (const float4*)iemb, (float4*)fI, 1.0f / 3.0f, ni4);

    const int*   E[2] = {e0, e1};
    const float* W[2] = {w0, w1};
    const int edgeBlocks = (NEDGE + (tpb / 32) - 1) / (tpb / 32);

    for (int t = 0; t < 2; ++t) {
        // zero u1,i1,u2,i2 (contiguous region) in one launch
        int nz4 = (int)((2 * (NU + NI)) / 4);
        zero_f4<<<(nz4 + tpb - 1) / tpb, tpb, 0, stream>>>((float4*)ws, nz4);

        // layer 1: embeddings -> (u1,i1)
        scatter_layer<<<edgeBlocks, tpb, 0, stream>>>(uemb, iemb, u1, i1,
                                                      E[t], W[t], NEDGE);
        // layer 2: (u1,i1) -> (u2,i2)
        scatter_layer<<<edgeBlocks, tpb, 0, stream>>>(u1, i1, u2, i2,
                                                      E[t], W[t], NEDGE);

        // final += softmax(tw)[t]/3 * (L1 + L2)
        accum_layers<<<(nu4 + tpb - 1) / tpb, tpb, 0, stream>>>(
            (float4*)fU, (const float4*)u1, (const float4*)u2, tw, t, nu4);
        accum_layers<<<(ni4 + tpb - 1) / tpb, tpb, 0, stream>>>(
            (float4*)fI, (const float4*)i1, (const float4*)i2, tw, t, ni4);
    }

    // 16 dots per wave, 8 waves per block -> 128 batch rows per block
    dot_wmma<<<BATCH / 128, tpb, 0, stream>>>(fU, fI, uidx, iidx, out);
}